// DeformConv2D_26680336843162
// MI455X (gfx1250) — compile-verified
//
#include <hip/hip_runtime.h>

typedef __attribute__((ext_vector_type(2))) float v2f;
typedef __attribute__((ext_vector_type(8))) float v8f;

#define HH 96
#define WW 96
#define CIN 64
#define COUT 64
#define NPIX (HH * WW)            // 9216
#define NTAPS 9
#define KTOT (CIN * NTAPS)        // 576
#define TILE_P 128                // pixels per workgroup
#define TBL (NTAPS * TILE_P)      // 1152 (tap,pixel) table entries
#define CH_CHUNK 4                // channels per K chunk
#define KCHUNK (CH_CHUNK * NTAPS) // 36 K values per chunk (9 wmma steps)
#define BSTRIDE 37                // padded LDS row stride for B panel [TILE_P][KCHUNK]

__global__ __launch_bounds__(256)
void deform_conv_wmma_f32(const float* __restrict__ x,
                          const float* __restrict__ off,
                          const float* __restrict__ Wf,   // (64, 576) row-major == (64,64,3,3)
                          float* __restrict__ out) {
    __shared__ int   sIdx[4 * TBL];          // 18 KB: 4 corner indices per (tap,pixel)
    __shared__ float sWt [4 * TBL];          // 18 KB: 4 corner weights (0 if out of image)
    __shared__ float sB  [TILE_P * BSTRIDE]; // 18.5 KB: cols panel, [pixel][k_local]

    const int tile = blockIdx.x;   // 0..71 (pixel tile)
    const int b    = blockIdx.y;   // 0..7  (batch)
    const int tid  = threadIdx.x;
    const int lane = tid & 31;
    const int wave = tid >> 5;     // 0..7
    const int m_blk  = wave & 3;   // which 16 output channels
    const int n_half = wave >> 1 >> 1; // wave>>2: which 64-pixel half
    const int laneN  = lane & 15;
    const int kHi    = (lane >> 4) << 1;  // 0 for lanes 0-15, 2 for lanes 16-31

    // ---- Phase 0: bilinear sample table, shared across all 64 channels ----
    for (int e = tid; e < TBL; e += 256) {
        const int t = e >> 7;          // tap 0..8
        const int p = e & (TILE_P - 1);
        const int P = tile * TILE_P + p;
        const int i = P / WW, j = P % WW;
        const float ox = off[(((size_t)b * 18 + t) * HH + i) * WW + j];
        const float oy = off[(((size_t)b * 18 + 9 + t) * HH + i) * WW + j];
        const int dx = t / 3 - 1, dy = t % 3 - 1;
        float px = (float)(i + 1 + dx) + ox;
        float py = (float)(j + 1 + dy) + oy;
        const float fx = floorf(px), fy = floorf(py);
        // out-of-boundary mask: snap to floor (reference semantics), then clip
        if (px < 1.f || px > 96.f) px = fminf(fmaxf(fx, 0.f), 97.f);
        if (py < 1.f || py > 96.f) py = fminf(fmaxf(fy, 0.f), 97.f);
        const float qlx = fminf(fmaxf(fx,       0.f), 97.f);
        const float qly = fminf(fmaxf(fy,       0.f), 97.f);
        const float qrx = fminf(fmaxf(fx + 1.f, 0.f), 97.f);
        const float qry = fminf(fmaxf(fy + 1.f, 0.f), 97.f);
        const float glt = (1.f + (qlx - px)) * (1.f + (qly - py));
        const float grb = (1.f - (qrx - px)) * (1.f - (qry - py));
        const float glb = (1.f + (qlx - px)) * (1.f - (qry - py));
        const float grt = (1.f - (qrx - px)) * (1.f + (qly - py));
        const float cu[4] = {qlx, qrx, qlx, qrx};
        const float cv[4] = {qly, qry, qry, qly};
        const float cg[4] = {glt, grb, glb, grt};
#pragma unroll
        for (int c4 = 0; c4 < 4; ++c4) {
            const int iu = (int)cu[c4], iv = (int)cv[c4];
            // padded image is zero outside [1,96]^2 -> weight 0
            const bool in = (iu >= 1) && (iu <= HH) && (iv >= 1) && (iv <= WW);
            sIdx[c4 * TBL + e] = in ? ((iu - 1) * WW + (iv - 1)) : 0;
            sWt [c4 * TBL + e] = in ? cg[c4] : 0.f;
        }
    }

    const v8f vzero = {0.f, 0.f, 0.f, 0.f, 0.f, 0.f, 0.f, 0.f};
    v8f acc[4];
    acc[0] = vzero; acc[1] = vzero; acc[2] = vzero; acc[3] = vzero;

    const int rowA = m_blk * 16 + laneN;  // W row for this lane's A fragment

    for (int ch = 0; ch < CIN / CH_CHUNK; ++ch) {
        __syncthreads();  // table ready (iter 0) / previous wmma done reading sB

        // ---- gather + bilinear blend: 36 x 128 column values into LDS ----
        const float* xb0 = x + ((size_t)b * CIN + ch * CH_CHUNK) * NPIX;
        for (int it = 0; it < (KCHUNK * TILE_P) / 256; ++it) {  // 18 iters
            const int e  = tid + it * 256;
            const int p  = e & (TILE_P - 1);
            const int kt = e >> 7;         // 0..35
            const int cl = kt / NTAPS;
            const int t  = kt - cl * NTAPS;
            const int ent = t * TILE_P + p;
            const float* xb = xb0 + (size_t)cl * NPIX;
            float v = sWt[0 * TBL + ent] * xb[sIdx[0 * TBL + ent]]
                    + sWt[1 * TBL + ent] * xb[sIdx[1 * TBL + ent]]
                    + sWt[2 * TBL + ent] * xb[sIdx[2 * TBL + ent]]
                    + sWt[3 * TBL + ent] * xb[sIdx[3 * TBL + ent]];
            sB[p * BSTRIDE + kt] = v;
        }
        __syncthreads();

        // ---- 9 WMMA K-steps over this 36-wide K chunk ----
#pragma unroll
        for (int ks = 0; ks < KCHUNK / 4; ++ks) {
            const int kglob = ch * KCHUNK + ks * 4 + kHi;
            // A fragment per ISA 16x4 layout: lane<16 -> K=k0,k0+1; lane>=16 -> K=k0+2,k0+3
            const v2f a = *(const v2f*)(Wf + (size_t)rowA * KTOT + kglob);
            const int kloc = ks * 4 + kHi;
#pragma unroll
            for (int s = 0; s < 4; ++s) {
                const int n = (n_half * 4 + s) * 16 + laneN;
                v2f bb;
                bb.x = sB[n * BSTRIDE + kloc];
                bb.y = sB[n * BSTRIDE + kloc + 1];
                acc[s] = __builtin_amdgcn_wmma_f32_16x16x4_f32(
                    false, a, false, bb, (short)0, acc[s], false, false);
            }
        }
    }

    // ---- store: C/D layout = 8 VGPRs, lanes 0-15: M=r, lanes 16-31: M=r+8 ----
    const int Pbase = tile * TILE_P;
    const int mOfs  = m_blk * 16 + ((lane >> 4) << 3);
#pragma unroll
    for (int s = 0; s < 4; ++s) {
        const int P = Pbase + (n_half * 4 + s) * 16 + laneN;
#pragma unroll
        for (int r = 0; r < 8; ++r) {
            out[((size_t)b * COUT + (mOfs + r)) * NPIX + P] = acc[s][r];
        }
    }
}

extern "C" void kernel_launch(void* const* d_in, const int* in_sizes, int n_in,
                              void* d_out, int out_size, void* d_ws, size_t ws_size,
                              hipStream_t stream) {
    (void)in_sizes; (void)n_in; (void)out_size; (void)d_ws; (void)ws_size;
    const float* x   = (const float*)d_in[0];
    const float* off = (const float*)d_in[1];
    const float* Wf  = (const float*)d_in[2];
    float* out = (float*)d_out;
    dim3 grid(NPIX / TILE_P, 8);  // 72 pixel tiles x 8 batches
    deform_conv_wmma_f32<<<grid, dim3(256), 0, stream>>>(x, off, Wf, out);
}